// resgnn_block_5394478923808
// MI455X (gfx1250) — compile-verified
//
#include <hip/hip_runtime.h>

#define D 128

typedef __attribute__((ext_vector_type(2))) float v2f;
typedef __attribute__((ext_vector_type(8))) float v8f;

// ---------------- init: agg (=d_out) <- 0, deg <- 1.0 (self loop) ----------------
__global__ void resgnn_init_kernel(float* __restrict__ deg, float* __restrict__ agg,
                                   int N, long total) {
    long i = (long)blockIdx.x * blockDim.x + threadIdx.x;
    if (i < total) agg[i] = 0.0f;
    if (i < N)     deg[i] = 1.0f;
}

// ---------------- degree count over dst ----------------
__global__ void resgnn_deg_kernel(const long long* __restrict__ ei,
                                  float* __restrict__ deg, int E) {
    int e = blockIdx.x * blockDim.x + threadIdx.x;
    if (e >= E) return;
    int d = (int)ei[(long)E + e];
    atomicAdd(&deg[d], 1.0f);
}

// ---------------- dinv = rsqrt(deg), in place ----------------
__global__ void resgnn_rsqrt_kernel(float* __restrict__ deg, int N) {
    int i = blockIdx.x * blockDim.x + threadIdx.x;
    if (i >= N) return;
    float v = deg[i];
    deg[i] = (v > 0.0f) ? rsqrtf(v) : 0.0f;
}

// ---------------- h = x @ W via V_WMMA_F32_16X16X4_F32 ----------------
// One wave per 16-row tile; 8 waves (256 threads) per block; W staged in LDS.
__global__ void resgnn_gemm_kernel(const float* __restrict__ x,
                                   const float* __restrict__ W,
                                   float* __restrict__ h, int N, int tiles) {
    __shared__ float Ws[D * D];                 // 64 KB, row-major W[k][n]
    int tid = threadIdx.x;
    for (int i = tid; i < D * D; i += 256) Ws[i] = W[i];
    __syncthreads();

    int wave = tid >> 5;
    int lane = tid & 31;
    int tile = blockIdx.x * 8 + wave;
    if (tile >= tiles) return;                  // wave-uniform: EXEC stays all-ones

    int rowBase = tile * 16;
    int m    = lane & 15;
    int koff = (lane >> 4) << 1;                // lanes 0-15: K+{0,1}; 16-31: K+{2,3}
    int row  = rowBase + m;
    if (row >= N) row = N - 1;                  // clamped read; tail rows never stored
    const float* xrow = x + (size_t)row * D;

    // Preload all A fragments for K = 0..127 (32 steps of K=4)
    v2f afrag[32];
#pragma unroll
    for (int ks = 0; ks < 32; ++ks) {
        const float* p = xrow + ks * 4 + koff;  // contiguous pair -> b64 load
        afrag[ks].x = p[0];
        afrag[ks].y = p[1];
    }

    int n0   = lane & 15;
    int rAdd = (lane >> 4) * 8;                 // C/D rows: v (lanes 0-15), v+8 (16-31)
    for (int nt = 0; nt < 8; ++nt) {
        v8f acc = {0.f, 0.f, 0.f, 0.f, 0.f, 0.f, 0.f, 0.f};
        int n = nt * 16 + n0;
#pragma unroll
        for (int ks = 0; ks < 32; ++ks) {
            int kb = ks * 4 + koff;
            v2f bfrag;
            bfrag.x = Ws[kb * D + n];           // B: K=kb (=ks*4+koff) row, col n
            bfrag.y = Ws[(kb + 1) * D + n];     //    K=kb+1 row, col n
            acc = __builtin_amdgcn_wmma_f32_16x16x4_f32(
                false, afrag[ks], false, bfrag, (short)0, acc, false, false);
        }
#pragma unroll
        for (int v = 0; v < 8; ++v) {
            int r = rowBase + rAdd + v;
            if (r < N) h[(size_t)r * D + n] = acc[v];
        }
    }
}

// ---------------- scatter: agg[dst] += h[src] * dinv[src]*dinv[dst] ----------------
// One wave32 per edge; each lane handles a float4 chunk of the 128-wide feature.
__global__ void resgnn_scatter_kernel(const long long* __restrict__ ei,
                                      const float* __restrict__ dinv,
                                      const float* __restrict__ h,
                                      float* __restrict__ agg, int E) {
    long gid = (long)blockIdx.x * blockDim.x + threadIdx.x;
    int e    = (int)(gid >> 5);
    int lane = (int)(gid & 31);
    if (e >= E) return;
    int s = (int)ei[e];
    int d = (int)ei[(long)E + e];
    float norm = dinv[s] * dinv[d];
    const float4 hv = *(const float4*)(h + (size_t)s * D + lane * 4);
    float* ap = agg + (size_t)d * D + lane * 4;
    atomicAdd(ap + 0, hv.x * norm);
    atomicAdd(ap + 1, hv.y * norm);
    atomicAdd(ap + 2, hv.z * norm);
    atomicAdd(ap + 3, hv.w * norm);
}

// ---------------- finish: out = x + relu(agg + selfloop + bias) ----------------
__global__ void resgnn_finish_kernel(const float* __restrict__ x,
                                     const float* __restrict__ h,
                                     const float* __restrict__ dinv,
                                     const float* __restrict__ bias,
                                     float* __restrict__ out, long total) {
    long i = (long)blockIdx.x * blockDim.x + threadIdx.x;
    if (i >= total) return;
    int rowi = (int)(i >> 7);
    int f    = (int)(i & (D - 1));
    float di  = dinv[rowi];
    float agg = out[i] + h[i] * (di * di) + bias[f];
    out[i] = x[i] + fmaxf(agg, 0.0f);
}

extern "C" void kernel_launch(void* const* d_in, const int* in_sizes, int n_in,
                              void* d_out, int out_size, void* d_ws, size_t ws_size,
                              hipStream_t stream) {
    const float*     x  = (const float*)d_in[0];
    const long long* ei = (const long long*)d_in[1];   // int64 edge_index [2, E]
    const float*     W  = (const float*)d_in[2];
    const float*     b  = (const float*)d_in[3];
    float* out = (float*)d_out;

    int N = in_sizes[0] / D;
    int E = in_sizes[1] / 2;
    long total = (long)N * D;

    // workspace: dinv (N floats) | h (N*D floats); agg aliases d_out
    char*  ws   = (char*)d_ws;
    float* dinv = (float*)ws;
    size_t off  = ((size_t)N * sizeof(float) + 255) & ~(size_t)255;
    float* h    = (float*)(ws + off);

    {
        int blocks = (int)((total + 255) / 256);
        resgnn_init_kernel<<<blocks, 256, 0, stream>>>(dinv, out, N, total);
    }
    resgnn_deg_kernel<<<(E + 255) / 256, 256, 0, stream>>>(ei, dinv, E);
    resgnn_rsqrt_kernel<<<(N + 255) / 256, 256, 0, stream>>>(dinv, N);

    int tiles  = (N + 15) / 16;            // 6250 for N=100000 (exact)
    int gblocks = (tiles + 7) / 8;
    resgnn_gemm_kernel<<<gblocks, 256, 0, stream>>>(x, W, h, N, tiles);

    long sthreads = (long)E * 32;
    resgnn_scatter_kernel<<<(int)((sthreads + 255) / 256), 256, 0, stream>>>(ei, dinv, h, out, E);

    resgnn_finish_kernel<<<(int)((total + 255) / 256), 256, 0, stream>>>(x, h, dinv, b, out, total);
}